// L1NMF_Net_78778290143263
// MI455X (gfx1250) — compile-verified
//
#include <hip/hip_runtime.h>
#include <math.h>

typedef float v2f __attribute__((ext_vector_type(2)));
typedef float v8f __attribute__((ext_vector_type(8)));

#define WMMA_F32(a, b, c) \
  __builtin_amdgcn_wmma_f32_16x16x4_f32(false, (a), false, (b), (short)0, (c), false, false)

// ---------------------------------------------------------------------------
// Init: A = A0^T (A0 is [N,R] row-major), B = A. A has ARP row slots; rows
// R..ARP-1 are zeroed so WMMA B-fragments can read them unguarded.
// ---------------------------------------------------------------------------
template <int TR>
__global__ void k_init(const float* __restrict__ A0, float* __restrict__ A,
                       float* __restrict__ B, int R, int ARP, long N) {
  long n = (long)blockIdx.x * blockDim.x + threadIdx.x;
  if (n >= N) return;
#pragma unroll
  for (int r = 0; r < TR; ++r) {
    if (r >= R) break;
    float v = A0[n * R + r];
    A[(long)r * N + n] = v;
    B[(long)r * N + n] = v;
  }
  for (int r = R; r < ARP; ++r) A[(long)r * N + n] = 0.0f;
}

// ---------------------------------------------------------------------------
// C1 = Wk^T @ M   [R x R]
// ---------------------------------------------------------------------------
__global__ void k_c1(const float* __restrict__ Wk, const float* __restrict__ M,
                     float* __restrict__ C1, int bands, int R) {
  int t = threadIdx.x;
  if (t >= R * R) return;
  int r = t / R, q = t % R;
  float s = 0.f;
  for (int b = 0; b < bands; ++b) s += Wk[b * R + r] * M[b * R + q];
  C1[t] = s;
}

// ---------------------------------------------------------------------------
// WkPad[b*16 + m] = (m<R) ? Wk[b*R+m] : 0   -- zero-padded for guard-free WMMA
// ---------------------------------------------------------------------------
__global__ void k_wkpad(const float* __restrict__ Wk, float* __restrict__ WkPad,
                        int bands, int R) {
  int t = blockIdx.x * blockDim.x + threadIdx.x;
  if (t >= bands * 16) return;
  int b = t >> 4, m = t & 15;
  WkPad[t] = (m < R) ? Wk[b * R + m] : 0.0f;
}

// ---------------------------------------------------------------------------
// Per-row self_active constants: SC[r] = tau_, SC[R+r] = p*lam
// ---------------------------------------------------------------------------
__global__ void k_selfconsts(const float* __restrict__ lambd, const float* __restrict__ p_arr,
                             int k, int R, float* __restrict__ SC) {
  int r = threadIdx.x;
  if (r >= R) return;
  float p = p_arr[k];
  float lam = lambd[k * R + r];
  float e1 = 1.0f / (2.0f - p);
  float e2 = (p - 1.0f) / (2.0f - p);
  float tau_ = powf(2.0f * (1.0f - p) * lam, e1) +
               p * lam * powf(2.0f * lam * (1.0f - p), e2);
  SC[r] = tau_;
  SC[R + r] = p * lam;
}

// ---------------------------------------------------------------------------
// WX = Wk^T @ X   [R x N]  via V_WMMA_F32_16X16X4_F32.
// Each wave produces TWO 16-col D tiles (32 columns) sharing the Wk A-frag:
// two independent accumulator chains -> loads overlap WMMA.
// Fragment layout (32-bit): lane m = lane&15; lanes<16 hold K={t,t+1} in
// VGPR0/1, lanes>=16 hold K={t+2,t+3}.
// ---------------------------------------------------------------------------
__global__ void k_wx(const float* __restrict__ X, const float* __restrict__ WkPad,
                     float* __restrict__ WXo, int bands, int R, long N) {
  int lane = threadIdx.x & 31;
  int wave = threadIdx.x >> 5;
  long n0 = ((long)blockIdx.x * (blockDim.x >> 5) + wave) * 32;
  if (n0 >= N) return;                       // wave-uniform exit
  int m  = lane & 15;
  int hi = lane >> 4;

  if (n0 + 32 <= N && (bands & 3) == 0) {
    // ---- dual-tile fast path: no guards, shared A-fragment ----
    v8f acc0 = {}, acc1 = {};
    const float* wp  = WkPad + (2 * hi) * 16 + m;   // padded stride 16
    const float* xp0 = X + (long)(2 * hi) * N + n0 + m;
    const float* xp1 = xp0 + N;
    long xstep = 4 * N;
    int iters = bands >> 2;
#pragma unroll 2
    for (int it = 0; it < iters; ++it) {
      v2f a, b0, b1;
      a.x  = wp[0];
      a.y  = wp[16];
      b0.x = xp0[0];
      b0.y = xp1[0];
      b1.x = xp0[16];
      b1.y = xp1[16];
      acc0 = WMMA_F32(a, b0, acc0);
      acc1 = WMMA_F32(a, b1, acc1);
      wp += 64;
      xp0 += xstep;
      xp1 += xstep;
    }
#pragma unroll
    for (int v = 0; v < 8; ++v) {
      int row = v + 8 * hi;
      if (row < R) {
        WXo[(long)row * N + n0 + m]      = acc0[v];
        WXo[(long)row * N + n0 + 16 + m] = acc1[v];
      }
    }
  } else {
    // ---- generic guarded path (two 16-col sub-tiles) ----
    for (int sub = 0; sub < 32; sub += 16) {
      long nn = n0 + sub + m;
      bool nok = (nn < N);
      long ns = nok ? nn : (N - 1);
      v8f acc = {};
      for (int t = 0; t < bands; t += 4) {
        int kx = t + 2 * hi;
        bool k0 = (kx < bands), k1 = (kx + 1 < bands);
        int kxs0 = k0 ? kx : 0;
        int kxs1 = k1 ? (kx + 1) : 0;
        v2f a, b;
        a.x = k0 ? WkPad[kxs0 * 16 + m] : 0.0f;
        a.y = k1 ? WkPad[kxs1 * 16 + m] : 0.0f;
        b.x = (nok && k0) ? X[(long)kxs0 * N + ns] : 0.0f;
        b.y = (nok && k1) ? X[(long)kxs1 * N + ns] : 0.0f;
        acc = WMMA_F32(a, b, acc);
      }
#pragma unroll
      for (int v = 0; v < 8; ++v) {
        int row = v + 8 * hi;
        if (row < R && nok) WXo[(long)row * N + nn] = acc[v];
      }
    }
  }
}

// ---------------------------------------------------------------------------
// Split-K partials of  P @ A^T  where P is [prows x N], A is [R x N] padded
// to >=16 readable rows. Each wave covers TWO 16-row P tiles (m0, m0+16)
// sharing the A B-fragment loads. part[s][prows*R]; deterministic reduce.
// K-adjacent fragment elements are contiguous -> float2 (b64) loads.
// ---------------------------------------------------------------------------
__global__ void k_gemmT(const float* P, const float* A, float* __restrict__ part,
                        int prows, int prowsRead, int R, long N, int splits) {
  int lane = threadIdx.x & 31;
  int wave = threadIdx.x >> 5;
  int s = blockIdx.x * (blockDim.x >> 5) + wave;
  if (s >= splits) return;                   // wave-uniform
  int m0 = blockIdx.y * 32;
  long chunk = (N + splits - 1) / splits;
  long nb = (long)s * chunk;
  if (nb >= N) return;                       // wave-uniform
  long ne = nb + chunk; if (ne > N) ne = N;
  int mL = lane & 15;
  int hi = lane >> 4;
  int q = lane & 15;
  float* dst = part + (long)s * prows * R;
  bool fastK = ((chunk & 3) == 0) && (nb + chunk <= N);

  if (fastK && m0 + 32 <= prowsRead) {
    // ---- dual-tile fast path ----
    v8f acc0 = {}, acc1 = {};
    const float* pp0 = P + (long)(m0 + mL) * N + nb + 2 * hi;
    const float* pp1 = pp0 + (long)16 * N;
    const float* ap  = A + (long)q * N + nb + 2 * hi;
    long iters = chunk >> 2;
#pragma unroll 2
    for (long it = 0; it < iters; ++it) {
      v2f a0 = *(const v2f*)pp0;
      v2f a1 = *(const v2f*)pp1;
      v2f b  = *(const v2f*)ap;
      acc0 = WMMA_F32(a0, b, acc0);
      acc1 = WMMA_F32(a1, b, acc1);
      pp0 += 4;
      pp1 += 4;
      ap += 4;
    }
    if (q < R) {
#pragma unroll
      for (int v = 0; v < 8; ++v) {
        int mm0 = m0 + v + 8 * hi;
        int mm1 = mm0 + 16;
        if (mm0 < prows) dst[mm0 * R + q] = acc0[v];
        if (mm1 < prows) dst[mm1 * R + q] = acc1[v];
      }
    }
    return;
  }

  // ---- tile 0 ----
  if (fastK && m0 + 16 <= prowsRead) {
    v8f acc = {};
    const float* pp = P + (long)(m0 + mL) * N + nb + 2 * hi;
    const float* ap = A + (long)q * N + nb + 2 * hi;
    long iters = chunk >> 2;
#pragma unroll 2
    for (long it = 0; it < iters; ++it) {
      v2f a = *(const v2f*)pp;
      v2f b = *(const v2f*)ap;
      acc = WMMA_F32(a, b, acc);
      pp += 4;
      ap += 4;
    }
    if (q < R) {
#pragma unroll
      for (int v = 0; v < 8; ++v) {
        int mm = m0 + v + 8 * hi;
        if (mm < prows) dst[mm * R + q] = acc[v];
      }
    }
  } else {
    // generic guarded tile 0
    int m = m0 + mL;
    bool mok = (m < prowsRead);
    bool qok = (q < R);
    long mrow = (long)(mok ? m : 0) * N;
    long qrow = (long)(qok ? q : 0) * N;
    v8f acc = {};
    for (long n = nb; n < ne; n += 4) {
      long kx = n + 2 * hi;
      bool k0 = (kx < ne), k1 = (kx + 1 < ne);
      long kxa = k0 ? kx : nb;
      long kxb = k1 ? (kx + 1) : nb;
      v2f a, b;
      a.x = (mok && k0) ? P[mrow + kxa] : 0.0f;
      a.y = (mok && k1) ? P[mrow + kxb] : 0.0f;
      b.x = (qok && k0) ? A[qrow + kxa] : 0.0f;
      b.y = (qok && k1) ? A[qrow + kxb] : 0.0f;
      acc = WMMA_F32(a, b, acc);
    }
    if (qok) {
#pragma unroll
      for (int v = 0; v < 8; ++v) {
        int mm = m0 + v + 8 * hi;
        if (mm < prows) dst[mm * R + q] = acc[v];
      }
    }
  }

  // ---- tile 1 (generic; only if it holds valid output rows) ----
  if (m0 + 16 < prows) {
    int m = m0 + 16 + mL;
    bool mok = (m < prowsRead);
    bool qok = (q < R);
    long mrow = (long)(mok ? m : 0) * N;
    long qrow = (long)(qok ? q : 0) * N;
    v8f acc = {};
    for (long n = nb; n < ne; n += 4) {
      long kx = n + 2 * hi;
      bool k0 = (kx < ne), k1 = (kx + 1 < ne);
      long kxa = k0 ? kx : nb;
      long kxb = k1 ? (kx + 1) : nb;
      v2f a, b;
      a.x = (mok && k0) ? P[mrow + kxa] : 0.0f;
      a.y = (mok && k1) ? P[mrow + kxb] : 0.0f;
      b.x = (qok && k0) ? A[qrow + kxa] : 0.0f;
      b.y = (qok && k1) ? A[qrow + kxb] : 0.0f;
      acc = WMMA_F32(a, b, acc);
    }
    if (qok) {
#pragma unroll
      for (int v = 0; v < 8; ++v) {
        int mm = m0 + 16 + v + 8 * hi;
        if (mm < prows) dst[mm * R + q] = acc[v];
      }
    }
  }
}

__global__ void k_reduce(const float* __restrict__ part, float* __restrict__ out,
                         int elems, int splits) {
  int t = blockIdx.x * blockDim.x + threadIdx.x;
  if (t >= elems) return;
  float s = 0.f;
  for (int i = 0; i < splits; ++i) s += part[(long)i * elems + t];
  out[t] = s;
}

// ---------------------------------------------------------------------------
// update_A (fused): grad = C1@A - WX + alpha*(A-B); step; self_active; relu;
// sum2one. One thread per column n.
// ---------------------------------------------------------------------------
template <int TR>
__global__ void k_update_a(const float* __restrict__ A, const float* __restrict__ B,
                           const float* __restrict__ WXp, const float* __restrict__ C1,
                           const float* __restrict__ SC, const float* __restrict__ Lk,
                           const float* __restrict__ alphak, const float* __restrict__ pk,
                           int k, float* __restrict__ Aout, int R, long N) {
  __shared__ float sC1[TR * TR];
  __shared__ float sSC[2 * TR];
  int t = threadIdx.x;
  if (t < R * R) sC1[t] = C1[t];
  if (t < 2 * R) sSC[t] = SC[t];
  __syncthreads();
  long n = (long)blockIdx.x * blockDim.x + t;
  if (n >= N) return;
  float Lv = Lk[k], al = alphak[k], p = pk[k];
  float a[TR], v[TR];
#pragma unroll
  for (int r = 0; r < TR; ++r) a[r] = (r < R) ? A[(long)r * N + n] : 0.0f;
  float s = 0.f;
#pragma unroll
  for (int r = 0; r < TR; ++r) {
    if (r >= R) break;
    float g = 0.f;
#pragma unroll
    for (int q = 0; q < TR; ++q)
      if (q < R) g += sC1[r * R + q] * a[q];
    g += -WXp[(long)r * N + n] + al * (a[r] - B[(long)r * N + n]);
    float x = a[r] - Lv * g;
    float vv;
    if (x - sSC[r] > 0.0f) {
      float xs = fabsf(x);
      vv = copysignf(1.0f, x) * (xs - sSC[R + r] * powf(xs, p - 1.0f));
    } else {
      vv = 0.0f;
    }
    vv = fminf(vv, 1.0f);   // min(v,1)
    vv = fmaxf(vv, 0.0f);   // relu
    v[r] = vv;
    s += vv;
  }
  float inv = 1.0f / (s + 1e-4f);
#pragma unroll
  for (int r = 0; r < TR; ++r) {
    if (r >= R) break;
    Aout[(long)r * N + n] = v[r] * inv;
  }
}

// ---------------------------------------------------------------------------
// prox_TV (one projected-gradient step) fused with B = sum2one(relu(.)).
// Pure 7-point stencil: div_x(i) = rw(i)-rw(i-1) with rw==0 on padded edge.
// ---------------------------------------------------------------------------
template <int TR>
__global__ void k_prox(const float* __restrict__ Ain, float* __restrict__ Aout,
                       float* __restrict__ Bout, const int* __restrict__ Hd,
                       const int* __restrict__ Wd, const float* __restrict__ tauk,
                       const float* __restrict__ alphak, int k, int R, long N) {
  long idx = (long)blockIdx.x * blockDim.x + threadIdx.x;
  if (idx >= N) return;
  int Ww = *Wd, Hh = *Hd;
  int i = (int)(idx / Ww), j = (int)(idx % Ww);
  float lmb = 2.0f * tauk[k] / alphak[k];
  float inv8 = 1.0f / (8.0f * lmb);
  float vals[TR];
  float ssum = 0.f;
#pragma unroll
  for (int c = 0; c < TR; ++c) {
    if (c >= R) break;
    const float* b = Ain + (long)c * N;
    float b00 = b[idx];
    // rw, sw at (i,j)
    float r0 = (i < Hh - 1) ? (b00 - b[idx + Ww]) * inv8 : 0.0f;
    float s0 = (j < Ww - 1) ? (b00 - b[idx + 1]) * inv8 : 0.0f;
    float w0 = fmaxf(1.0f, sqrtf(r0 * r0 + s0 * s0 + 1e-8f));
    float div = (r0 + s0) / w0;
    if (i > 0) {  // rw at (i-1,j)
      float bm = b[idx - Ww];
      float r1 = (bm - b00) * inv8;
      float s1 = (j < Ww - 1) ? (bm - b[idx - Ww + 1]) * inv8 : 0.0f;
      float w1 = fmaxf(1.0f, sqrtf(r1 * r1 + s1 * s1 + 1e-8f));
      div -= r1 / w1;
    }
    if (j > 0) {  // sw at (i,j-1)
      float bl = b[idx - 1];
      float s2 = (bl - b00) * inv8;
      float r2 = (i < Hh - 1) ? (bl - b[idx - 1 + Ww]) * inv8 : 0.0f;
      float w2 = fmaxf(1.0f, sqrtf(r2 * r2 + s2 * s2 + 1e-8f));
      div -= s2 / w2;
    }
    float val = b00 - lmb * div;
    Aout[(long)c * N + idx] = val;
    float z = fmaxf(val, 0.0f);
    vals[c] = z;
    ssum += z;
  }
  float inv = 1.0f / (ssum + 1e-4f);
#pragma unroll
  for (int c = 0; c < TR; ++c) {
    if (c >= R) break;
    Bout[(long)c * N + idx] = vals[c] * inv;
  }
}

// ---------------------------------------------------------------------------
// M_new = relu(M - L2*(M@AA - XAT))
// ---------------------------------------------------------------------------
__global__ void k_update_m(const float* __restrict__ Mi, const float* __restrict__ AA,
                           const float* __restrict__ XAT, const float* __restrict__ L2k,
                           int k, float* __restrict__ Mo, int bands, int R) {
  int t = blockIdx.x * blockDim.x + threadIdx.x;
  if (t >= bands * R) return;
  int b = t / R, q = t % R;
  float g = 0.f;
  for (int r = 0; r < R; ++r) g += Mi[b * R + r] * AA[r * R + q];
  g -= XAT[t];
  Mo[t] = fmaxf(0.0f, Mi[t] - L2k[k] * g);
}

// ---------------------------------------------------------------------------
extern "C" void kernel_launch(void* const* d_in, const int* in_sizes, int n_in,
                              void* d_out, int out_size, void* d_ws, size_t ws_size,
                              hipStream_t stream) {
  const float* X     = (const float*)d_in[0];
  const float* M0    = (const float*)d_in[1];
  const float* A0    = (const float*)d_in[2];
  const float* Wa    = (const float*)d_in[3];
  const float* lambd = (const float*)d_in[4];
  const float* L     = (const float*)d_in[5];
  const float* L2    = (const float*)d_in[6];
  const float* p     = (const float*)d_in[7];
  const float* alpha = (const float*)d_in[8];
  const float* tau   = (const float*)d_in[9];
  const int*   Hd    = (const int*)d_in[10];
  const int*   Wd    = (const int*)d_in[11];

  int K = in_sizes[5];                 // L is [K]
  int R = in_sizes[4] / K;             // lambd is [K,1,R]
  int bands = in_sizes[1] / R;         // M0 is [bands,R]
  long N = (long)in_sizes[2] / R;      // A0 is [N,R]
  int ARP = (R <= 16) ? 16 : R;        // padded row slots for the A buffer

  const int SPL = 512;

  float* ws = (float*)d_ws;
  size_t o = 0;
  float* A_cur = ws + o; o += (size_t)ARP * N;   // padded: rows R..ARP-1 zero
  float* A_nxt = ws + o; o += (size_t)R * N;
  float* B_cur = ws + o; o += (size_t)R * N;
  float* WX    = ws + o; o += (size_t)R * N;
  float* Mb0   = ws + o; o += (size_t)bands * R;
  float* Mb1   = ws + o; o += (size_t)bands * R;
  float* C1    = ws + o; o += (size_t)R * R;
  float* AA    = ws + o; o += (size_t)R * R;
  float* XAT   = ws + o; o += (size_t)bands * R;
  float* SC    = ws + o; o += (size_t)2 * R;
  float* WkPad = ws + o; o += (size_t)bands * 16;
  float* PART  = ws + o; o += (size_t)SPL * bands * R;
  float* PART2 = ws + o; o += (size_t)SPL * R * R;
  if (o * sizeof(float) > ws_size) return;  // workspace too small (shouldn't happen)

  int nb256 = (int)((N + 255) / 256);
  bool r8 = (R == 8);

  if (r8) k_init<8><<<nb256, 256, 0, stream>>>(A0, A_cur, B_cur, R, ARP, N);
  else    k_init<16><<<nb256, 256, 0, stream>>>(A0, A_cur, B_cur, R, ARP, N);
  hipMemcpyAsync(Mb0, M0, (size_t)bands * R * sizeof(float),
                 hipMemcpyDeviceToDevice, stream);

  float* Mc = Mb0;
  float* Mn = Mb1;
  for (int k = 0; k < K; ++k) {
    const float* Wk = Wa + (size_t)k * bands * R;
    k_c1<<<1, 256, 0, stream>>>(Wk, Mc, C1, bands, R);
    k_wkpad<<<(bands * 16 + 255) / 256, 256, 0, stream>>>(Wk, WkPad, bands, R);
    k_selfconsts<<<1, 64, 0, stream>>>(lambd, p, k, R, SC);

    {  // WX = Wk^T @ X  (WMMA, one full streaming pass over X)
      long tiles = (N + 31) / 32;            // 32 cols per wave
      int blocks = (int)((tiles + 3) / 4);   // 4 waves per block
      k_wx<<<blocks, 128, 0, stream>>>(X, WkPad, WX, bands, R, N);
    }

    if (r8) k_update_a<8><<<nb256, 256, 0, stream>>>(A_cur, B_cur, WX, C1, SC, L,
                                                     alpha, p, k, A_nxt, R, N);
    else    k_update_a<16><<<nb256, 256, 0, stream>>>(A_cur, B_cur, WX, C1, SC, L,
                                                      alpha, p, k, A_nxt, R, N);

    if (r8) k_prox<8><<<nb256, 256, 0, stream>>>(A_nxt, A_cur, B_cur, Hd, Wd, tau,
                                                 alpha, k, R, N);
    else    k_prox<16><<<nb256, 256, 0, stream>>>(A_nxt, A_cur, B_cur, Hd, Wd, tau,
                                                  alpha, k, R, N);

    // AA = A @ A^T (WMMA split-K, deterministic reduce); reads padded rows
    k_gemmT<<<dim3(SPL / 4, (R + 31) / 32), 128, 0, stream>>>(A_cur, A_cur, PART2,
                                                              R, ARP, R, N, SPL);
    k_reduce<<<(R * R + 255) / 256, 256, 0, stream>>>(PART2, AA, R * R, SPL);

    // XAT = X @ A^T (WMMA split-K, second streaming pass over X)
    k_gemmT<<<dim3(SPL / 4, (bands + 31) / 32), 128, 0, stream>>>(X, A_cur, PART,
                                                                  bands, bands, R, N, SPL);
    k_reduce<<<(bands * R + 255) / 256, 256, 0, stream>>>(PART, XAT, bands * R, SPL);

    k_update_m<<<(bands * R + 255) / 256, 256, 0, stream>>>(Mc, AA, XAT, L2, k, Mn,
                                                            bands, R);
    float* tmp = Mc; Mc = Mn; Mn = tmp;
  }

  hipMemcpyAsync(d_out, Mc, (size_t)bands * R * sizeof(float),
                 hipMemcpyDeviceToDevice, stream);
  hipMemcpyAsync((float*)d_out + (size_t)bands * R, A_cur,
                 (size_t)R * N * sizeof(float), hipMemcpyDeviceToDevice, stream);
}